// Stiefel_12919261626682
// MI455X (gfx1250) — compile-verified
//
#include <hip/hip_runtime.h>
#include <hip/hip_bf16.h>

// Problem constants (batch derived at launch from in_sizes).
#define DIMN   256
#define RANK   64
#define NMIN   (DIMN - RANK)                 // 192
#define N0     (DIMN*RANK - RANK*(RANK+1)/2) // 14304
#define THETA_FLOATS (2*N0)                  // 28608 floats per batch row
#define COL_STRIDE 257                       // float2 per column: stride 514 dwords
                                             // -> banks {2k,2k+1} per lane, conflict-free

// LDS: ret tile only: 64 cols x 257 float2 = 131584 B  -> 2 workgroups / WGP,
// 4 wave32s across the 4 SIMD32s (round 1's 246KB stranded half the WGP).
#define LDS_BYTES (RANK*COL_STRIDE*8)

__device__ __forceinline__ int group_off(int j) {
    // offsets[j] = sum_{m<j} (NMIN+m) = NMIN*j + j*(j-1)/2   (in (t,p) pairs)
    return NMIN*j + ((j*(j-1)) >> 1);
}

__device__ __forceinline__ unsigned lds_offset_of(const void* p) {
    // generic pointer to LDS: low 32 bits = workgroup-relative LDS byte offset
    return (unsigned)(unsigned long long)p;
}

__global__ __launch_bounds__(64)
void stiefel_euler_kernel(const float* __restrict__ theta,
                          float* __restrict__ out)
{
    extern __shared__ float smem[];
    float2* sRet = (float2*)smem;             // 64 x 257 float2

    const int tid = threadIdx.x;              // 0..63 == creation index k
    const int b   = blockIdx.x;
    const float2* gTheta = (const float2*)(theta + (size_t)b * THETA_FLOATS);

    const int k  = tid;
    const int Nk = NMIN + k;                  // rowJ_k uses group k (size Nk)
    float2* col  = sRet + (size_t)k * COL_STRIDE;

    // ---- rowJ_k: col[n] = ct_n * cumprod(st)_{n-1} * exp(i*(sum_{m<n} p_m - p_n))
    {
        const float2* g = gTheta + group_off(k);
        float s = 0.0f, cp = 1.0f;
        for (int n = 0; n < Nk; ++n) {
            float2 tp = g[n];                 // (t, p)
            float ph  = s - tp.y;
            float v   = __cosf(tp.x) * cp;
            col[n] = make_float2(v * __cosf(ph), v * __sinf(ph));
            s  += tp.y;
            cp *= __sinf(tp.x);
        }
        col[Nk] = make_float2(cp * __cosf(s), cp * __sinf(s));
    }

    // ---- Transforms T_{k+1}..T_{63}: sequential complex Givens chain,
    // column updated in place in LDS, grows by one element per transform.
    // Theta streams from global (whole array is L2-resident: 117MB < 192MB L2).
    for (int j = k + 1; j < RANK; ++j) {
        const float2* g = gTheta + group_off(j);
        __builtin_prefetch(g, 0, 1);          // global_prefetch_b8: warm the stream
        const int L = NMIN + j;
        float ax = 0.0f, ay = 0.0f;           // complex accumulator a
        #pragma unroll 2
        for (int i = 0; i < L; ++i) {
            float2 tp = g[i];
            float ct = __cosf(tp.x), st = __sinf(tp.x);
            float ex = __cosf(tp.y), ey = __sinf(tp.y);
            float2 bv = col[i];
            // z = (ct*a - st*b) * conj(e^{ip})
            float zrx = ct*ax - st*bv.x;
            float zry = ct*ay - st*bv.y;
            col[i] = make_float2(zrx*ex + zry*ey, zry*ex - zrx*ey);
            // a = (ct*b + st*a) * e^{ip}
            float arx = ct*bv.x + st*ax;
            float ary = ct*bv.y + st*ay;
            ax = arx*ex - ary*ey;
            ay = ary*ex + arx*ey;
        }
        col[L] = make_float2(ax, ay);
    }

    // ---- Writeback via CDNA5 async LDS->global stores (1 instruction per
    // element, no VGPR roundtrip). Lanes reconverge here, and creation index
    // k maps to output column RANK-1-k (innermost dim) -> the 32 lanes of a
    // wave store 32 contiguous 8B elements per iteration.
    asm volatile("s_wait_dscnt 0x0" ::: "memory");   // async engine must see ds_stores
    const int outc = RANK - 1 - k;
    float2* outp = (float2*)out + (size_t)b * DIMN * RANK + outc;
    #pragma unroll 4
    for (int i = 0; i < DIMN; ++i) {
        unsigned ldsOff = lds_offset_of(col + i);
        const float2* gdst = outp + (size_t)i * RANK;
        asm volatile("global_store_async_from_lds_b64 %0, %1, off"
                     :: "v"(gdst), "v"(ldsOff) : "memory");
    }
    asm volatile("s_wait_asynccnt 0x0" ::: "memory");
}

extern "C" void kernel_launch(void* const* d_in, const int* in_sizes, int n_in,
                              void* d_out, int out_size, void* d_ws, size_t ws_size,
                              hipStream_t stream) {
    const float* theta = (const float*)d_in[0];
    float* out = (float*)d_out;               // complex64 -> interleaved f32
    const int batch = in_sizes[0] / THETA_FLOATS;
    stiefel_euler_kernel<<<dim3(batch), dim3(64), LDS_BYTES, stream>>>(theta, out);
}